// dispModel_27436251087434
// MI455X (gfx1250) — compile-verified
//
#include <hip/hip_runtime.h>
#include <hip/hip_bf16.h>
#include <math.h>

// Problem constants (from reference): B=32, L=2048, M=99 modes, H=8 hidden.
#define BDIM   32
#define LDIM   2048
#define MDIM   99
#define OUTL   1848   // L - 200
#define TILES_TOTAL 116          // 116*16 >= 1848 positions
#define TILE_CHUNKS 4            // gridDim.y split: 4 chunks of 29 tiles
#define TILES_PER_CHUNK (TILES_TOTAL / TILE_CHUNKS)

typedef float v2f __attribute__((ext_vector_type(2)));
typedef float v8f __attribute__((ext_vector_type(8)));

// ---------------------------------------------------------------------------
// Kernel 1: per-row exclusive-prefix cumsum.  cs[row][0]=0, cs[row][i+1]=sum x[0..i].
// rows 0..31 = bias, rows 32..63 = atac (matches x = concat([bias, atac])).
// ---------------------------------------------------------------------------
__global__ __launch_bounds__(256)
void cumsum_rows_kernel(const float* __restrict__ atac,
                        const float* __restrict__ bias,
                        float* __restrict__ cs)
{
    const int rowid = blockIdx.x;
    const float* x = (rowid < BDIM) ? (bias + rowid * LDIM)
                                    : (atac + (rowid - BDIM) * LDIM);
    float* csr = cs + rowid * (LDIM + 1);

    __shared__ float part[256];
    const int tid = threadIdx.x;

    float v[8];
    float run = 0.f;
#pragma unroll
    for (int k = 0; k < 8; ++k) { run += x[tid * 8 + k]; v[k] = run; }
    part[tid] = run;
    __syncthreads();
    if (tid == 0) {                       // tiny serial scan of 256 partials
        float acc = 0.f;
        for (int i = 0; i < 256; ++i) { float t = part[i]; part[i] = acc; acc += t; }
    }
    __syncthreads();
    const float pre = part[tid];
#pragma unroll
    for (int k = 0; k < 8; ++k) csr[tid * 8 + k + 1] = pre + v[k];
    if (tid == 0) csr[0] = 0.f;
}

// ---------------------------------------------------------------------------
// Kernel 2: fuse the two affine pairs per mode.
//   W12p[m] : 6x8  rows 0..4 = S1@L1, row 5 = s1b@L1 + l1b   (const-1 feature)
//   W34 [m] : 8x4  = L2@S2
//   B34 [m] : 4    = l2b@S2 + s2b
// ---------------------------------------------------------------------------
__global__ __launch_bounds__(128)
void combine_weights_kernel(const float* __restrict__ s1w, const float* __restrict__ s1b,
                            const float* __restrict__ l1w, const float* __restrict__ l1b,
                            const float* __restrict__ l2w, const float* __restrict__ l2b,
                            const float* __restrict__ s2w, const float* __restrict__ s2b,
                            float* __restrict__ w12p, float* __restrict__ w34,
                            float* __restrict__ b34)
{
    const int m   = blockIdx.x;
    const int tid = threadIdx.x;
    const float* S1  = s1w + m * 25;
    const float* S1b = s1b + m * 5;
    const float* L1  = l1w + m * 40;
    const float* L1b = l1b + m * 8;
    const float* L2  = l2w + m * 32;
    const float* L2b = l2b + m * 4;
    const float* S2  = s2w + m * 16;
    const float* S2b = s2b + m * 4;

    if (tid < 48) {                       // W12p (6x8)
        const int r = tid / 8, g = tid % 8;
        float acc;
        if (r < 5) {
            acc = 0.f;
            for (int k = 0; k < 5; ++k) acc += S1[r * 5 + k] * L1[k * 8 + g];
        } else {
            acc = L1b[g];
            for (int k = 0; k < 5; ++k) acc += S1b[k] * L1[k * 8 + g];
        }
        w12p[m * 48 + r * 8 + g] = acc;
    } else if (tid < 80) {                // W34 (8x4)
        const int i = tid - 48, h = i / 4, o = i % 4;
        float acc = 0.f;
        for (int k = 0; k < 4; ++k) acc += L2[h * 4 + k] * S2[k * 4 + o];
        w34[m * 32 + h * 4 + o] = acc;
    } else if (tid < 84) {                // b34 (4)
        const int o = tid - 80;
        float acc = S2b[o];
        for (int k = 0; k < 4; ++k) acc += L2b[k] * S2[k * 4 + o];
        b34[m * 4 + o] = acc;
    }
}

// ---------------------------------------------------------------------------
// Kernel 3: fused window-sum + MLP (f32 WMMA) + z-score epilogue.
// One wave32 per (b, mode, tile-chunk); each tile = 16 output positions.
//   Stage 1: D1(16x16) = A1(W12'^T, 16x6 padded) x B1(fg^T, 6x16)     [2x wmma K=4]
//   Stage 2: D2(16x16) = A2(W34^T,  16x8 padded) x B2(h^T,  8x16)+b34 [2x wmma K=4]
// A frag (ISA 7.12.2, 32-bit 16x4): lane<16 -> K={0,1}, lane>=16 -> K={2,3}, row=lane&15.
// B frag: lane n -> column n; halves split K the same way.
// D/C frag: VGPR r, lane -> element (r + 8*(lane>=16), lane&15).
// ---------------------------------------------------------------------------
__global__ __launch_bounds__(32)
void fused_disp_kernel(const float* __restrict__ cs,
                       const float* __restrict__ w12p,
                       const float* __restrict__ w34,
                       const float* __restrict__ b34,
                       const int*   __restrict__ modes,
                       float* __restrict__ out)
{
    const int pair = blockIdx.x;
    const int b  = pair / MDIM;
    const int mi = pair % MDIM;
    const int lane = threadIdx.x;
    const int row  = lane & 15;
    const bool hi  = lane >= 16;

    const int m = modes[mi];
    const int s = m >> 1;

    const float* csb = cs + b * (LDIM + 1);            // bias row
    const float* csa = cs + (BDIM + b) * (LDIM + 1);   // atac row
    const float* W12 = w12p + mi * 48;                 // [k(6)][i(8)]
    const float* W34 = w34  + mi * 32;                 // [k(8)][i(4)]
    const float* B34 = b34  + mi * 4;

    // ---- A fragments (per-mode constants, hoisted out of the tile loop) ----
    const int kb = hi ? 2 : 0;
    v2f a1c0, a1c1, a2c0, a2c1;
    a1c0.x = (row < 8) ? W12[(kb + 0) * 8 + row] : 0.f;
    a1c0.y = (row < 8) ? W12[(kb + 1) * 8 + row] : 0.f;
    a1c1.x = (!hi && row < 8) ? W12[4 * 8 + row] : 0.f;   // K=4
    a1c1.y = (!hi && row < 8) ? W12[5 * 8 + row] : 0.f;   // K=5 (bias row, feature==1)
    a2c0.x = (row < 4) ? W34[(kb + 0) * 4 + row] : 0.f;
    a2c0.y = (row < 4) ? W34[(kb + 1) * 4 + row] : 0.f;
    a2c1.x = (row < 4) ? W34[(kb + 4) * 4 + row] : 0.f;
    a2c1.y = (row < 4) ? W34[(kb + 5) * 4 + row] : 0.f;

    // C accumulator for stage 2 carries b34 (rows 0..3 only, broadcast over columns)
    v8f c2 = {};
    c2[0] = hi ? 0.f : B34[0];
    c2[1] = hi ? 0.f : B34[1];
    c2[2] = hi ? 0.f : B34[2];
    c2[3] = hi ? 0.f : B34[3];

    const int base_out = (b * MDIM + mi) * OUTL;
    const int t0 = blockIdx.y * TILES_PER_CHUNK;

    for (int t = t0; t < t0 + TILES_PER_CHUNK; ++t) {
        const int j = 100 + t * 16 + row;

        int i0 = j + 1 - m - 2 * s; i0 = i0 < 0 ? 0 : (i0 > LDIM ? LDIM : i0);
        int i1 = j + 1 - m;         i1 = i1 < 0 ? 0 : (i1 > LDIM ? LDIM : i1);
        int i2 = j + 1 + m;         i2 = i2 < 0 ? 0 : (i2 > LDIM ? LDIM : i2);
        int i3 = j + 1 + m + 2 * s; i3 = i3 < 0 ? 0 : (i3 > LDIM ? LDIM : i3);

        const float cb0 = csb[i0], cb1 = csb[i1], cb2 = csb[i2], cb3 = csb[i3];
        const float ca0 = csa[i0], ca1 = csa[i1], ca2 = csa[i2], ca3 = csa[i3];
        const float bl = cb1 - cb0, bc = cb2 - cb1, br = cb3 - cb2;
        const float il = ca1 - ca0, ic = ca2 - ca1, ir = ca3 - ca2;
        const float lt = ic + il, rt = ic + ir;
        const float lt10 = log10f(lt), rt10 = log10f(rt);

        // B1 = fg^T: features [bl, br, bc, log10(lt) | log10(rt), 1, 0, 0]
        v2f b1c0, b1c1;
        b1c0.x = hi ? bc   : bl;
        b1c0.y = hi ? lt10 : br;
        b1c1.x = hi ? 0.f  : rt10;
        b1c1.y = hi ? 0.f  : 1.0f;

        v8f zero = {};
        v8f d1 = __builtin_amdgcn_wmma_f32_16x16x4_f32(
            false, a1c0, false, b1c0, (short)0, zero, false, false);
        d1 = __builtin_amdgcn_wmma_f32_16x16x4_f32(
            false, a1c1, false, b1c1, (short)0, d1, false, false);
#pragma unroll
        for (int r = 0; r < 8; ++r) d1[r] = fmaxf(d1[r], 0.f);   // ReLU

        // Re-stage h^T rows 2,3,6,7 (live in lanes 0..15) into the upper-lane B halves.
        const float h2 = __shfl(d1[2], row, 32);
        const float h3 = __shfl(d1[3], row, 32);
        const float h6 = __shfl(d1[6], row, 32);
        const float h7 = __shfl(d1[7], row, 32);
        v2f b2c0, b2c1;
        b2c0.x = hi ? h2 : d1[0];
        b2c0.y = hi ? h3 : d1[1];
        b2c1.x = hi ? h6 : d1[4];
        b2c1.y = hi ? h7 : d1[5];

        v8f d2 = __builtin_amdgcn_wmma_f32_16x16x4_f32(
            false, a2c0, false, b2c0, (short)0, c2, false, false);
        d2 = __builtin_amdgcn_wmma_f32_16x16x4_f32(
            false, a2c1, false, b2c1, (short)0, d2, false, false);

        // Lanes 0..15: D2 vgprs 0..3 hold (lpm, lps, rpm, rps) for position j.
        const float lpm = d2[0];
        const float lps = fmaxf(d2[1], 0.f);
        const float rpm = d2[2];
        const float rps = fmaxf(d2[3], 0.f);
        const float lps_s = (lps == 0.f) ? 1.f : lps;
        const float rps_s = (rps == 0.f) ? 1.f : rps;
        const float lz = (ic / lt - lpm) / lps_s;
        const float rz = (ic / rt - rpm) / rps_s;
        float z = fmaxf(lz, rz);
        const bool bad = (lt < 1.f) || (rt < 1.f) || (lps == 0.f) || (rps == 0.f);
        if (bad) z = __builtin_nanf("");
        if (__builtin_isinf(z)) z = __builtin_nanf("");
        if (!__builtin_isnan(z)) z = fminf(fmaxf(z, -10.f), 10.f);

        if (!hi && j < 100 + OUTL) out[base_out + (j - 100)] = z;
    }
}

// ---------------------------------------------------------------------------
// Launch
// ---------------------------------------------------------------------------
extern "C" void kernel_launch(void* const* d_in, const int* in_sizes, int n_in,
                              void* d_out, int out_size, void* d_ws, size_t ws_size,
                              hipStream_t stream)
{
    const float* atac = (const float*)d_in[0];
    const float* bias = (const float*)d_in[1];
    const float* s1w  = (const float*)d_in[2];
    const float* s1b  = (const float*)d_in[3];
    const float* l1w  = (const float*)d_in[4];
    const float* l1b  = (const float*)d_in[5];
    const float* l2w  = (const float*)d_in[6];
    const float* l2b  = (const float*)d_in[7];
    const float* s2w  = (const float*)d_in[8];
    const float* s2b  = (const float*)d_in[9];
    const int*   modes = (const int*)d_in[10];

    float* ws   = (float*)d_ws;
    float* cs   = ws;                           // 64 * 2049
    float* w12p = cs + 64 * (LDIM + 1);         // 99 * 48
    float* w34  = w12p + MDIM * 48;             // 99 * 32
    float* b34  = w34  + MDIM * 32;             // 99 * 4

    cumsum_rows_kernel<<<2 * BDIM, 256, 0, stream>>>(atac, bias, cs);
    combine_weights_kernel<<<MDIM, 128, 0, stream>>>(s1w, s1b, l1w, l1b,
                                                     l2w, l2b, s2w, s2b,
                                                     w12p, w34, b34);
    fused_disp_kernel<<<dim3(BDIM * MDIM, TILE_CHUNKS), 32, 0, stream>>>(
        cs, w12p, w34, b34, modes, (float*)d_out);
}